// WeaveLayer_42485816492561
// MI455X (gfx1250) — compile-verified
//
#include <hip/hip_runtime.h>

typedef __attribute__((ext_vector_type(2))) float v2f;
typedef __attribute__((ext_vector_type(8))) float v8f;

#define N_NODES 100000
#define N_EDGES 400000
#define D 128
#define BN_EPS 1e-5f

// ---------------- ws layout (floats) ----------------
// [0..127]    column sums
// [128..255]  column sum-of-squares
// [256..383]  a[c] = gamma*rsqrt(var+eps)
// [384..511]  b[c] = beta - a*mu
// [512.. )    h buffer, N_NODES*128 floats (51.2 MB)
#define WS_STATS 0
#define WS_AB    256
#define WS_H     512

__global__ void zero_stats_kernel(float* ws) {
    ws[threadIdx.x] = 0.0f;   // 256 threads zero sum+sumsq
}

// 256 threads: r = t>>5 (8 rows/iter), cb = t&31 (4 cols each)
__global__ void bn_stats_kernel(const float* __restrict__ x, float* __restrict__ ws, int N) {
    __shared__ float ls[8 * 32 * 4];
    __shared__ float lq[8 * 32 * 4];
    const int t = threadIdx.x;
    const int r = t >> 5;
    const int cb = t & 31;
    float4 s = make_float4(0.f, 0.f, 0.f, 0.f);
    float4 q = make_float4(0.f, 0.f, 0.f, 0.f);
    for (int row = blockIdx.x * 8 + r; row < N; row += gridDim.x * 8) {
        const float4 xv = *(const float4*)(x + (size_t)row * D + cb * 4);
        s.x += xv.x; s.y += xv.y; s.z += xv.z; s.w += xv.w;
        q.x += xv.x * xv.x; q.y += xv.y * xv.y; q.z += xv.z * xv.z; q.w += xv.w * xv.w;
    }
    *(float4*)(ls + (r * 32 + cb) * 4) = s;
    *(float4*)(lq + (r * 32 + cb) * 4) = q;
    __syncthreads();
    if (r == 0) {
        float4 S = make_float4(0.f, 0.f, 0.f, 0.f);
        float4 Q = make_float4(0.f, 0.f, 0.f, 0.f);
        #pragma unroll
        for (int i = 0; i < 8; ++i) {
            const float4 a = *(const float4*)(ls + (i * 32 + cb) * 4);
            const float4 c = *(const float4*)(lq + (i * 32 + cb) * 4);
            S.x += a.x; S.y += a.y; S.z += a.z; S.w += a.w;
            Q.x += c.x; Q.y += c.y; Q.z += c.z; Q.w += c.w;
        }
        atomicAdd(&ws[WS_STATS + cb * 4 + 0], S.x);
        atomicAdd(&ws[WS_STATS + cb * 4 + 1], S.y);
        atomicAdd(&ws[WS_STATS + cb * 4 + 2], S.z);
        atomicAdd(&ws[WS_STATS + cb * 4 + 3], S.w);
        atomicAdd(&ws[WS_STATS + 128 + cb * 4 + 0], Q.x);
        atomicAdd(&ws[WS_STATS + 128 + cb * 4 + 1], Q.y);
        atomicAdd(&ws[WS_STATS + 128 + cb * 4 + 2], Q.z);
        atomicAdd(&ws[WS_STATS + 128 + cb * 4 + 3], Q.w);
    }
}

__global__ void bn_finalize_kernel(const float* __restrict__ gamma,
                                   const float* __restrict__ beta,
                                   float* __restrict__ ws, float invN) {
    const int c = threadIdx.x;  // 128 threads
    const float mu  = ws[WS_STATS + c] * invN;
    const float var = ws[WS_STATS + 128 + c] * invN - mu * mu;
    const float rs  = rsqrtf(var + BN_EPS);
    const float a   = gamma[c] * rs;
    ws[WS_AB + c]       = a;
    ws[WS_AB + 128 + c] = beta[c] - a * mu;
}

// One block = 16 rows. 256 threads = 8 waves; wave j computes out cols [16j,16j+16).
// h = relu(a*x+b) @ W^T via v_wmma_f32_16x16x4_f32, K-loop of 32 steps.
__global__ void __launch_bounds__(256)
gemm_bn_relu_kernel(const float* __restrict__ x, const float* __restrict__ W,
                    const float* __restrict__ ws, float* __restrict__ h,
                    float* __restrict__ out) {
    __shared__ float ytile[16 * 132];  // row stride 132 -> conflict-free ds_load_b64
    const float* __restrict__ acoef = ws + WS_AB;
    const float* __restrict__ bcoef = ws + WS_AB + 128;
    const int row0 = blockIdx.x * 16;
    const int t = threadIdx.x;
    // ---- stage normalized tile: r = t>>5 (rows r, r+8), cb = t&31 (4 cols) ----
    {
        const int r = t >> 5;
        const int cb = t & 31;
        const float4 av = *(const float4*)(acoef + cb * 4);
        const float4 bv = *(const float4*)(bcoef + cb * 4);
        #pragma unroll
        for (int i = 0; i < 2; ++i) {
            const int rr = r + 8 * i;
            const float4 xv = *(const float4*)(x + (size_t)(row0 + rr) * D + cb * 4);
            float4 yv;
            yv.x = fmaxf(av.x * xv.x + bv.x, 0.f);
            yv.y = fmaxf(av.y * xv.y + bv.y, 0.f);
            yv.z = fmaxf(av.z * xv.z + bv.z, 0.f);
            yv.w = fmaxf(av.w * xv.w + bv.w, 0.f);
            *(float4*)(ytile + rr * 132 + cb * 4) = yv;
        }
    }
    __syncthreads();
    // ---- WMMA: wave owns 16x16 output tile ----
    const int wave = t >> 5;
    const int lane = t & 31;
    const int o0 = wave * 16;
    const int n = lane & 15;        // B/C/D column within tile
    const int half = lane >> 4;     // lane group
    const float* __restrict__ wrow = W + (size_t)(o0 + n) * D;  // W[o][k], row-major
    const float* __restrict__ yrow = ytile + (lane & 15) * 132; // A row (M = lane&15)
    v8f acc = {};
    #pragma unroll
    for (int kk = 0; kk < 32; ++kk) {
        const int k = kk * 4 + 2 * half;           // consecutive K pair
        const v2f afrag = *(const v2f*)(yrow + k); // ds_load_b64
        const v2f bfrag = *(const v2f*)(wrow + k); // global_load_b64 (L2-resident W)
        acc = __builtin_amdgcn_wmma_f32_16x16x4_f32(
            /*neg_a=*/false, afrag, /*neg_b=*/false, bfrag,
            /*c_mod=*/(short)0, acc, /*reuse_a=*/false, /*reuse_b=*/false);
    }
    // ---- write back: D vgpr r -> M = r + 8*half (ISA C/D layout) ----
    #pragma unroll
    for (int r = 0; r < 8; ++r) {
        const int m = r + 8 * half;
        const size_t idx = (size_t)(row0 + m) * D + o0 + n;
        const float v = acc[r];
        h[idx] = v;
        out[idx] = v * v;   // initializes d_out before edge scatter
    }
}

// One wave per edge: float4 gather of h[col] (L2 hit), 4x global_atomic_add_f32 into out[row].
__global__ void edge_scatter_kernel(const int* __restrict__ ei,
                                    const float* __restrict__ ew,
                                    const float* __restrict__ h,
                                    float* __restrict__ out, int E) {
    const int t = threadIdx.x;
    const int lane = t & 31;
    const int e = blockIdx.x * 8 + (t >> 5);
    if (e >= E) return;
    const int row = ei[e];       // edge_index[0][e]
    const int col = ei[E + e];   // edge_index[1][e]
    const float w = ew[e];
    const float4 hv = *(const float4*)(h + (size_t)col * D + lane * 4);
    float* op = out + (size_t)row * D + lane * 4;
    atomicAdd(op + 0, w * hv.x);
    atomicAdd(op + 1, w * hv.y);
    atomicAdd(op + 2, w * hv.z);
    atomicAdd(op + 3, w * hv.w);
}

extern "C" void kernel_launch(void* const* d_in, const int* in_sizes, int n_in,
                              void* d_out, int out_size, void* d_ws, size_t ws_size,
                              hipStream_t stream) {
    const float* n_feat = (const float*)d_in[0];
    const int*   eidx   = (const int*)d_in[1];
    const float* ewgt   = (const float*)d_in[2];
    const float* gamma  = (const float*)d_in[3];
    const float* beta   = (const float*)d_in[4];
    const float* W      = (const float*)d_in[5];
    float* out = (float*)d_out;
    float* ws  = (float*)d_ws;
    float* h   = ws + WS_H;

    zero_stats_kernel<<<1, 256, 0, stream>>>(ws);
    bn_stats_kernel<<<1024, 256, 0, stream>>>(n_feat, ws, N_NODES);
    bn_finalize_kernel<<<1, 128, 0, stream>>>(gamma, beta, ws, 1.0f / (float)N_NODES);
    gemm_bn_relu_kernel<<<N_NODES / 16, 256, 0, stream>>>(n_feat, W, ws, h, out);
    edge_scatter_kernel<<<(N_EDGES + 7) / 8, 256, 0, stream>>>(eidx, ewgt, h, out, N_EDGES);
}